// YOLOXHeadLoss_49185965473945
// MI455X (gfx1250) — compile-verified
//
#include <hip/hip_runtime.h>

// ---------------------------------------------------------------------------
// YOLOX SimOTA head loss for MI455X (gfx1250, wave32).
//
// CDNA5 feature use:
//  * V_WMMA_F32_16X16X4_F32: the SimOTA class-cost gather
//        cls_gather[n,g] = logit(p)[n, gt_labels[g]]
//    computed as an exact f32 GEMM against a one-hot matrix (20 K-steps for
//    C=80, 2 G-tiles for G=32, one wave per 16 anchors).
//  * ds_bpermute-based wave32 shuffles for the sum(log1mp) reduction and the
//    butterfly top-k merge.
//  * global_prefetch_b8 via __builtin_prefetch in the streaming top-k scan.
// ---------------------------------------------------------------------------

#define BB   16
#define CC   80
#define GG   32
#define NN   8400
#define TOPKK 10
#define BN   (BB * NN)   // 134400

typedef float v2f __attribute__((ext_vector_type(2)));
typedef float v8f __attribute__((ext_vector_type(8)));

__device__ __forceinline__ void anchor_info(int n, int& p, int& W, int& hw,
                                            float& s) {
  bool l0 = n < 6400, l1 = n < 8000;
  p  = l0 ? n : (l1 ? n - 6400 : n - 8000);
  W  = l0 ? 80 : (l1 ? 40 : 20);
  hw = l0 ? 6400 : (l1 ? 1600 : 400);
  s  = l0 ? 8.f : (l1 ? 16.f : 32.f);
}
// branch-free row/col with constant-divisor strength reduction
__device__ __forceinline__ void anchor_xy(int n, int p, int W, int& x, int& y) {
  bool l0 = n < 6400, l1 = n < 8000;
  y = l0 ? (p / 80) : (l1 ? (p / 40) : (p / 20));
  x = p - y * W;
}

__device__ __forceinline__ float sigmoidf_(float x) {
  return 1.f / (1.f + expf(-x));
}
__device__ __forceinline__ float bce_(float x, float t) {
  return fmaxf(x, 0.f) - x * t + log1pf(expf(-fabsf(x)));
}
__device__ __forceinline__ float iou_(float ax0, float ay0, float ax1, float ay1,
                                      float bx0, float by0, float bx1, float by1) {
  float tlx = fmaxf(ax0, bx0), tly = fmaxf(ay0, by0);
  float brx = fminf(ax1, bx1), bry = fminf(ay1, by1);
  float w = fmaxf(brx - tlx, 0.f), h = fmaxf(bry - tly, 0.f);
  float inter = w * h;
  float a1 = (ax1 - ax0) * (ay1 - ay0);
  float a2 = (bx1 - bx0) * (by1 - by0);
  return inter / fmaxf(a1 + a2 - inter, 1e-12f);
}

// Branch-free register-resident top-k insertions (constant indices only, so
// the 10-entry lists stay in VGPRs; dynamic indexing would spill to scratch).
__device__ __forceinline__ void ins_desc(float v, float (&tv)[TOPKK]) {
#pragma unroll
  for (int j = TOPKK - 1; j >= 1; --j) {
    bool a = v > tv[j];
    bool b = v > tv[j - 1];
    tv[j] = a ? (b ? tv[j - 1] : v) : tv[j];
  }
  tv[0] = (v > tv[0]) ? v : tv[0];
}
__device__ __forceinline__ bool keylt_(float c1, int i1, float c2, int i2) {
  return (c1 < c2) || (c1 == c2 && i1 < i2);   // jax top_k tie-break: low idx
}
__device__ __forceinline__ void ins_asc(float c, int i, float (&cv)[TOPKK],
                                        int (&ci)[TOPKK]) {
#pragma unroll
  for (int j = TOPKK - 1; j >= 1; --j) {
    bool a = keylt_(c, i, cv[j], ci[j]);
    bool b = keylt_(c, i, cv[j - 1], ci[j - 1]);
    float nc = a ? (b ? cv[j - 1] : c) : cv[j];
    int   ni = a ? (b ? ci[j - 1] : i) : ci[j];
    cv[j] = nc; ci[j] = ni;
  }
  bool a0 = keylt_(c, i, cv[0], ci[0]);
  cv[0] = a0 ? c : cv[0];
  ci[0] = a0 ? i : ci[0];
}

// ---------------------------------------------------------------------------
// Kernel 1: decode boxes for all B*N anchors.
// ---------------------------------------------------------------------------
__global__ __launch_bounds__(256) void k_decode(
    const float* __restrict__ r0, const float* __restrict__ r1,
    const float* __restrict__ r2, float* __restrict__ boxes) {
  int i = blockIdx.x * blockDim.x + threadIdx.x;
  if (i >= BN) return;
  int b = i / NN, n = i % NN;
  int p, W, hw; float s;
  anchor_info(n, p, W, hw, s);
  const float* rg = (n < 6400 ? r0 : (n < 8000 ? r1 : r2)) +
                    (size_t)b * 4 * hw + p;
  float dx = rg[0], dy = rg[(size_t)hw];
  float dw = rg[(size_t)2 * hw], dh = rg[(size_t)3 * hw];
  int xi, yi; anchor_xy(n, p, W, xi, yi);
  float px = (float)xi * s, py = (float)yi * s;
  float cx = dx * s + px, cy = dy * s + py;
  float w = expf(dw) * s, h = expf(dh) * s;
  float* o = boxes + (size_t)i * 4;
  o[0] = cx - 0.5f * w; o[1] = cy - 0.5f * h;
  o[2] = cx + 0.5f * w; o[3] = cy + 0.5f * h;
}

// ---------------------------------------------------------------------------
// Kernel 2: one wave per (image, 16-anchor tile). WMMA one-hot gather GEMM +
// sum(log1mp) shuffle reduction + pairwise IoU / center prior + cost matrix.
// Outputs cost/iou transposed: [b][g][n] for the per-GT top-k kernel.
// ---------------------------------------------------------------------------
__global__ __launch_bounds__(32) void k_cost(
    const float* __restrict__ cls0, const float* __restrict__ cls1,
    const float* __restrict__ cls2,
    const float* __restrict__ obj0, const float* __restrict__ obj1,
    const float* __restrict__ obj2,
    const float* __restrict__ boxes,  // [B][N][4]
    const float* __restrict__ gtb,    // [B][G][4]
    const int*   __restrict__ glab,   // [B][G]
    float* __restrict__ costT,        // [B][G][N]
    float* __restrict__ iouT) {       // [B][G][N]
  const int MT   = NN / 16;           // 525 tiles
  int blk  = blockIdx.x;
  int b    = blk / MT;
  int base = (blk % MT) * 16;
  int lane = threadIdx.x;
  int half = lane >> 4;               // 0: K=0/1, 1: K=2/3 (32-bit A layout)
  int mrow = lane & 15;

  // --- A operand source: anchor (base+mrow), class pair depends on half ---
  int m_a = base + mrow;
  int p, W, hw; float s;
  anchor_info(m_a, p, W, hw, s);
  const float* clsP = (m_a < 6400 ? cls0 : (m_a < 8000 ? cls1 : cls2)) +
                      (size_t)b * CC * hw + p;
  const float* objP = (m_a < 6400 ? obj0 : (m_a < 8000 ? obj1 : obj2)) +
                      (size_t)b * hw + p;
  float so = sigmoidf_(objP[0]);

  // one-hot B columns: lane's GT pair (mrow, mrow+16)
  int lblL = glab[b * GG + mrow];
  int lblH = glab[b * GG + 16 + mrow];

  v8f acc0 = {}; v8f acc1 = {};
  float partial = 0.f;  // per-lane partial of sum_c log(1-p) (half the classes)

  for (int kc = 0; kc < CC / 4; ++kc) {
    int c0 = kc * 4 + 2 * half;
    float x0 = clsP[(size_t)c0 * hw];
    float x1 = clsP[(size_t)(c0 + 1) * hw];
    float p0 = sqrtf(sigmoidf_(x0) * so);
    float p1 = sqrtf(sigmoidf_(x1) * so);
    p0 = fminf(fmaxf(p0, 1e-7f), 1.f - 1e-7f);
    p1 = fminf(fmaxf(p1, 1e-7f), 1.f - 1e-7f);
    float l0 = log1pf(-p0), l1 = log1pf(-p1);
    float r0 = logf(p0) - l0, r1 = logf(p1) - l1;  // logit(p)
    partial += l0 + l1;
    v2f A  = {r0, r1};
    v2f B0 = {lblL == c0 ? 1.f : 0.f, lblL == c0 + 1 ? 1.f : 0.f};
    v2f B1 = {lblH == c0 ? 1.f : 0.f, lblH == c0 + 1 ? 1.f : 0.f};
    acc0 = __builtin_amdgcn_wmma_f32_16x16x4_f32(false, A, false, B0,
                                                 (short)0, acc0, false, false);
    acc1 = __builtin_amdgcn_wmma_f32_16x16x4_f32(false, A, false, B1,
                                                 (short)0, acc1, false, false);
  }
  // full per-anchor sum_log1mp: lane m and lane m+16 cover complementary K sets
  float sumfull = partial + __shfl_xor(partial, 16, 32);

  // GT geometry for this lane's two GT columns
  const float* g0p = gtb + ((size_t)b * GG + mrow) * 4;
  const float* g1p = gtb + ((size_t)b * GG + 16 + mrow) * 4;
  float g0x0 = g0p[0], g0y0 = g0p[1], g0x1 = g0p[2], g0y1 = g0p[3];
  float g1x0 = g1p[0], g1y0 = g1p[1], g1x1 = g1p[2], g1y1 = g1p[3];
  float gc0x = 0.5f * (g0x0 + g0x1), gc0y = 0.5f * (g0y0 + g0y1);
  float gc1x = 0.5f * (g1x0 + g1x1), gc1y = 0.5f * (g1y0 + g1y1);

#pragma unroll
  for (int r = 0; r < 8; ++r) {
    // D-matrix layout: VGPR r holds M = r (lanes 0-15) / r+8 (lanes 16-31)
    int m_d    = r + 8 * half;
    int anchor = base + m_d;
    int p2, W2, hw2; float s2;
    anchor_info(anchor, p2, W2, hw2, s2);
    int xi, yi; anchor_xy(anchor, p2, W2, xi, yi);
    float cx = (float)xi * s2 + 0.5f * s2;
    float cy = (float)yi * s2 + 0.5f * s2;
    float rad = 2.5f * s2;
    const float* bx = boxes + ((size_t)b * NN + anchor) * 4;
    float b0 = bx[0], b1 = bx[1], b2 = bx[2], b3 = bx[3];

    float i0 = iou_(b0, b1, b2, b3, g0x0, g0y0, g0x1, g0y1);
    float i1 = iou_(b0, b1, b2, b3, g1x0, g1y0, g1x1, g1y1);
    bool inG0 = (cx > g0x0) && (cy > g0y0) && (cx < g0x1) && (cy < g0y1);
    bool inG1 = (cx > g1x0) && (cy > g1y0) && (cx < g1x1) && (cy < g1y1);
    bool inC0 = (cx > gc0x - rad) && (cy > gc0y - rad) &&
                (cx < gc0x + rad) && (cy < gc0y + rad);
    bool inC1 = (cx > gc1x - rad) && (cy > gc1y - rad) &&
                (cx < gc1x + rad) && (cy < gc1y + rad);

    // valid[anchor] = any_g(in_gt | in_ct); low 16 lanes cover anchor base+r,
    // high 16 lanes cover base+r+8, each lane contributing 2 GT columns.
    unsigned m = (unsigned)__ballot((inG0 | inC0) | (inG1 | inC1));
    bool valid = half ? ((m >> 16) != 0u) : ((m & 0xFFFFu) != 0u);

    float sm  = __shfl(sumfull, m_d, 32);
    float cc0 = -acc0[r] - sm;   // -(logp - log1mp)[lbl] - sum(log1mp)
    float cc1 = -acc1[r] - sm;
    float cost0 = valid ? (cc0 - 3.f * logf(i0 + 1e-7f) +
                           100000.f * ((inG0 && inC0) ? 0.f : 1.f)) : 1e8f;
    float cost1 = valid ? (cc1 - 3.f * logf(i1 + 1e-7f) +
                           100000.f * ((inG1 && inC1) ? 0.f : 1.f)) : 1e8f;
    size_t o0 = ((size_t)b * GG + mrow) * NN + anchor;
    size_t o1 = ((size_t)b * GG + 16 + mrow) * NN + anchor;
    costT[o0] = cost0; iouT[o0] = valid ? i0 : 0.f;
    costT[o1] = cost1; iouT[o1] = valid ? i1 : 0.f;
  }
}

// ---------------------------------------------------------------------------
// Kernel 3: one WAVE per (b,g). Lane-local top-10 over a strided slice
// (coalesced streaming loads + global_prefetch), then a 5-round shfl_xor
// butterfly merge so every lane holds the global top-10. dyn_k from top-10
// IoU sum; lane 0 atomicOrs bit g into the per-anchor match mask.
// ---------------------------------------------------------------------------
__global__ __launch_bounds__(32) void k_topk(
    const float* __restrict__ costT, const float* __restrict__ iouT,
    unsigned* __restrict__ Mbits) {
  int t = blockIdx.x;                  // (b,g) pair
  int b = t / GG, g = t % GG;
  int lane = threadIdx.x;
  const float* irow = iouT + (size_t)t * NN;
  const float* crow = costT + (size_t)t * NN;

  float tv[TOPKK];
  float cv[TOPKK]; int ci[TOPKK];
#pragma unroll
  for (int k = 0; k < TOPKK; ++k) {
    tv[k] = -1.f; cv[k] = 3.4e38f; ci[k] = 0x7fffffff;
  }

  for (int n = lane; n < NN; n += 32) {
    __builtin_prefetch(irow + n + 1024, 0, 0);
    __builtin_prefetch(crow + n + 1024, 0, 0);
    float v = irow[n];
    float c = crow[n];
    if (v > tv[TOPKK - 1]) ins_desc(v, tv);
    if (c <= cv[TOPKK - 1]) ins_asc(c, n, cv, ci);
  }

  // butterfly merge: after 5 rounds every lane holds the global top-10
  for (int d = 1; d < 32; d <<= 1) {
#pragma unroll
    for (int k = 0; k < TOPKK; ++k) {
      float pv = __shfl_xor(tv[k], d, 32);
      float pc = __shfl_xor(cv[k], d, 32);
      int   pi = __shfl_xor(ci[k], d, 32);
      ins_desc(pv, tv);
      ins_asc(pc, pi, cv, ci);
    }
  }

  float sk = 0.f;
#pragma unroll
  for (int k = 0; k < TOPKK; ++k) sk += tv[k];
  int dynk = (int)sk;                  // trunc toward zero (astype int32)
  dynk = dynk < 1 ? 1 : (dynk > TOPKK ? TOPKK : dynk);

  if (lane == 0) {
    unsigned bit = 1u << g;
#pragma unroll
    for (int k = 0; k < TOPKK; ++k)
      if (k < dynk) atomicOr(&Mbits[(size_t)b * NN + ci[k]], bit);
  }
}

// ---------------------------------------------------------------------------
// Kernel 4: per-anchor conflict resolution (argmin cost among matched GTs).
// assigned = -1 marks background.
// ---------------------------------------------------------------------------
__global__ __launch_bounds__(256) void k_resolve(
    const float* __restrict__ costT, const float* __restrict__ iouT,
    const unsigned* __restrict__ Mbits,
    int* __restrict__ assigned, float* __restrict__ posiou) {
  int i = blockIdx.x * blockDim.x + threadIdx.x;
  if (i >= BN) return;
  int b = i / NN, n = i % NN;
  unsigned bits = Mbits[i];
  if (!bits) { assigned[i] = -1; posiou[i] = 0.f; return; }
  int g;
  if (__popc(bits) > 1) {
    float best = 3.4e38f; int bg = 0;
    for (int q = 0; q < GG; ++q)
      if (bits & (1u << q)) {
        float c = costT[((size_t)b * GG + q) * NN + n];
        if (c < best) { best = c; bg = q; }     // ties -> lowest g (argmin)
      }
    g = bg;
  } else {
    g = __ffs(bits) - 1;
  }
  assigned[i] = g;
  posiou[i] = iouT[((size_t)b * GG + g) * NN + n];
}

// ---------------------------------------------------------------------------
// Kernel 5: loss partials + wave/LDS block reduction into ws accumulators
// acc = {sum_cls, sum_bbox, sum_obj, num_fg}.
// ---------------------------------------------------------------------------
__global__ __launch_bounds__(256) void k_loss(
    const float* __restrict__ cls0, const float* __restrict__ cls1,
    const float* __restrict__ cls2,
    const float* __restrict__ obj0, const float* __restrict__ obj1,
    const float* __restrict__ obj2,
    const float* __restrict__ boxes, const float* __restrict__ gtb,
    const int* __restrict__ glab,
    const int* __restrict__ assigned, const float* __restrict__ posiou,
    float* __restrict__ acc) {
  int i = blockIdx.x * blockDim.x + threadIdx.x;
  float lcls = 0.f, lbox = 0.f, lobj = 0.f, nfg = 0.f;
  if (i < BN) {
    int b = i / NN, n = i % NN;
    int p, W, hw; float s;
    anchor_info(n, p, W, hw, s);
    const float* objP = (n < 6400 ? obj0 : (n < 8000 ? obj1 : obj2)) +
                        (size_t)b * hw + p;
    int g = assigned[i];
    float fg = (g >= 0) ? 1.f : 0.f;
    lobj = bce_(objP[0], fg);
    if (g >= 0) {
      nfg = 1.f;
      const float* clsP = (n < 6400 ? cls0 : (n < 8000 ? cls1 : cls2)) +
                          (size_t)b * CC * hw + p;
      int lbl  = glab[b * GG + g];
      float pi = posiou[i];
      for (int c = 0; c < CC; ++c) {
        float t = (c == lbl) ? pi : 0.f;
        lcls += bce_(clsP[(size_t)c * hw], t);
      }
      const float* bx = boxes + (size_t)i * 4;
      const float* gb = gtb + ((size_t)b * GG + g) * 4;
      float iou = iou_(bx[0], bx[1], bx[2], bx[3], gb[0], gb[1], gb[2], gb[3]);
      iou = fmaxf(iou, 1e-16f);
      lbox = 1.f - iou * iou;
    }
  }
  __shared__ float red[4][8];
  int lane = threadIdx.x & 31, wid = threadIdx.x >> 5;
  float v[4] = {lcls, lbox, lobj, nfg};
#pragma unroll
  for (int k = 0; k < 4; ++k) {
    float x = v[k];
    for (int off = 16; off; off >>= 1) x += __shfl_down(x, off, 32);
    if (lane == 0) red[k][wid] = x;
  }
  __syncthreads();
  if (threadIdx.x == 0) {
    float s0 = 0.f, s1 = 0.f, s2 = 0.f, s3 = 0.f;
    for (int w = 0; w < 8; ++w) {
      s0 += red[0][w]; s1 += red[1][w]; s2 += red[2][w]; s3 += red[3][w];
    }
    atomicAdd(&acc[0], s0); atomicAdd(&acc[1], s1);
    atomicAdd(&acc[2], s2); atomicAdd(&acc[3], s3);
  }
}

// ---------------------------------------------------------------------------
// Kernel 6: normalize and write the 3 outputs (rewritten every launch).
// ---------------------------------------------------------------------------
__global__ void k_final(const float* __restrict__ acc, float* __restrict__ out) {
  if (threadIdx.x == 0 && blockIdx.x == 0) {
    float nt = fmaxf(acc[3], 1.f);
    out[0] = acc[0] / nt;
    out[1] = acc[1] / nt * 5.f;
    out[2] = acc[2] / nt;
  }
}

// ---------------------------------------------------------------------------
// Inputs (setup_inputs dict order):
//   0:cls0 1:reg0 2:obj0 3:cls1 4:reg1 5:obj1 6:cls2 7:reg2 8:obj2
//   9:gt_boxes 10:gt_labels(int32)
// Workspace layout (floats): [acc(4) pad to 64 | Mbits(BN u32) | assigned(BN)
//   | posiou(BN) | boxes(4BN) | costT(32BN) | iouT(32BN)]  ~= 36.4 MiB.
// ---------------------------------------------------------------------------
extern "C" void kernel_launch(void* const* d_in, const int* in_sizes, int n_in,
                              void* d_out, int out_size, void* d_ws,
                              size_t ws_size, hipStream_t stream) {
  const float* cls0 = (const float*)d_in[0];
  const float* reg0 = (const float*)d_in[1];
  const float* obj0 = (const float*)d_in[2];
  const float* cls1 = (const float*)d_in[3];
  const float* reg1 = (const float*)d_in[4];
  const float* obj1 = (const float*)d_in[5];
  const float* cls2 = (const float*)d_in[6];
  const float* reg2 = (const float*)d_in[7];
  const float* obj2 = (const float*)d_in[8];
  const float* gtb  = (const float*)d_in[9];
  const int*   glab = (const int*)d_in[10];
  float* out = (float*)d_out;

  float*    W        = (float*)d_ws;
  float*    acc      = W;
  unsigned* Mbits    = (unsigned*)(W + 64);
  int*      assigned = (int*)(W + 64 + (size_t)BN);
  float*    posiou   = W + 64 + 2 * (size_t)BN;
  float*    boxes    = W + 64 + 3 * (size_t)BN;
  float*    costT    = W + 64 + 7 * (size_t)BN;
  float*    iouT     = W + 64 + 39 * (size_t)BN;

  // zero accumulators + match masks each launch (graph-capturable)
  hipMemsetAsync(W, 0, (64 + (size_t)BN) * sizeof(float), stream);

  k_decode<<<BN / 256, 256, 0, stream>>>(reg0, reg1, reg2, boxes);
  k_cost<<<BB * (NN / 16), 32, 0, stream>>>(cls0, cls1, cls2, obj0, obj1, obj2,
                                            boxes, gtb, glab, costT, iouT);
  k_topk<<<BB * GG, 32, 0, stream>>>(costT, iouT, Mbits);
  k_resolve<<<BN / 256, 256, 0, stream>>>(costT, iouT, Mbits, assigned, posiou);
  k_loss<<<BN / 256, 256, 0, stream>>>(cls0, cls1, cls2, obj0, obj1, obj2,
                                       boxes, gtb, glab, assigned, posiou, acc);
  k_final<<<1, 32, 0, stream>>>(acc, out);
}